// ForwardSumLoss_2508260901108
// MI455X (gfx1250) — compile-verified
//
#include <hip/hip_runtime.h>
#include <cstdint>

#define BLANK_LP (-1.0f)
#define NEGF     (-1e30f)

// Problem shape (from reference setup_inputs)
#define TT  2000   // mel frames
#define LL  400    // text length
#define SS  801    // 2*LL + 1 CTC states
#define NTH 832    // 26 wave32s: covers all 801 states
#define PF  8      // async prefetch depth (rows ahead), power of two

// ---------- CDNA5 async global->LDS helpers (guarded, with sync fallback) ----
#if defined(__gfx1250__) && __has_builtin(__builtin_amdgcn_global_load_async_to_lds_b128)
#define HAVE_ASYNC_LDS 1
#else
#define HAVE_ASYNC_LDS 0
#endif

typedef int v4i __attribute__((ext_vector_type(4)));
typedef __attribute__((address_space(1))) v4i v4i_g;   // global
typedef __attribute__((address_space(3))) v4i v4i_l;   // LDS

__device__ __forceinline__ void async_cp16(const float* g, float* l) {
#if HAVE_ASYNC_LDS
  __builtin_amdgcn_global_load_async_to_lds_b128(
      (v4i_g*)(uintptr_t)g,   // integer reinterpret: global AS1 == flat address
      (v4i_l*)l,              // explicit generic->LDS addrspace cast
      0, 0);
#else
  *(float4*)l = *(const float4*)g;   // synchronous fallback
#endif
}

template <int N>
__device__ __forceinline__ void wait_async() {
#if HAVE_ASYNC_LDS
#if __has_builtin(__builtin_amdgcn_s_wait_asynccnt)
  __builtin_amdgcn_s_wait_asynccnt(N);
#else
  asm volatile("s_wait_asynccnt %0" ::"i"(N) : "memory");
#endif
#endif
}

// ---------- Kernel 1: logZ(b,t) = logsumexp(-1.0, attn[b,t,0..L-1]) ---------
// One wave32 per row; float4 loads; __shfl_xor butterfly reductions (wave32).
__global__ __launch_bounds__(256) void logz_kernel(const float* __restrict__ attn,
                                                   float* __restrict__ logz,
                                                   int nrows) {
  int row  = blockIdx.x * 8 + (threadIdx.x >> 5);
  if (row >= nrows) return;
  int lane = threadIdx.x & 31;

  const float4* p4 = (const float4*)(attn + (size_t)row * LL);  // 100 float4 / row
  float4 v0 = p4[lane];
  float4 v1 = p4[lane + 32];
  float4 v2 = p4[lane + 64];
  float4 v3 = (lane < 4) ? p4[lane + 96] : make_float4(NEGF, NEGF, NEGF, NEGF);

  float m = BLANK_LP;  // include the blank logit in the max
  m = fmaxf(m, fmaxf(fmaxf(v0.x, v0.y), fmaxf(v0.z, v0.w)));
  m = fmaxf(m, fmaxf(fmaxf(v1.x, v1.y), fmaxf(v1.z, v1.w)));
  m = fmaxf(m, fmaxf(fmaxf(v2.x, v2.y), fmaxf(v2.z, v2.w)));
  m = fmaxf(m, fmaxf(fmaxf(v3.x, v3.y), fmaxf(v3.z, v3.w)));
#pragma unroll
  for (int i = 16; i >= 1; i >>= 1) m = fmaxf(m, __shfl_xor(m, i, 32));

  float sum = (lane == 0) ? __expf(BLANK_LP - m) : 0.0f;
  sum += __expf(v0.x - m) + __expf(v0.y - m) + __expf(v0.z - m) + __expf(v0.w - m);
  sum += __expf(v1.x - m) + __expf(v1.y - m) + __expf(v1.z - m) + __expf(v1.w - m);
  sum += __expf(v2.x - m) + __expf(v2.y - m) + __expf(v2.z - m) + __expf(v2.w - m);
  sum += __expf(v3.x - m) + __expf(v3.y - m) + __expf(v3.z - m) + __expf(v3.w - m);
#pragma unroll
  for (int i = 16; i >= 1; i >>= 1) sum += __shfl_xor(sum, i, 32);

  if (lane == 0) logz[row] = m + __logf(sum);
}

// ---------- Kernel 2: CTC forward scan, one workgroup per batch row ---------
// State s lives on thread s; alpha double-buffered in LDS with a 2-entry NEG
// pad so alpha[s-1]/alpha[s-2] reads are branch-free. Logit rows stream in via
// an async global->LDS ring PF rows deep; s_wait_asynccnt retires the oldest
// transfer just before the single per-step barrier.
__global__ __launch_bounds__(NTH) void ctc_scan_kernel(const float* __restrict__ attn,
                                                       const int* __restrict__ text_lens,
                                                       const int* __restrict__ mel_lens,
                                                       const float* __restrict__ logz,
                                                       float* __restrict__ loss_b) {
  __shared__ __align__(16) float rowbuf[PF][LL];   // 12.8 KB logit-row ring
  __shared__ __align__(16) float zbuf[TT];         // 8 KB: logZ(b, :)
  __shared__ __align__(16) float alphaA[SS + 3];   // phys[0..1] = NEG pad
  __shared__ __align__(16) float alphaB[SS + 3];

  const int    tid     = threadIdx.x;
  const int    b       = blockIdx.x;
  const int    tl      = text_lens[b];
  const int    ml      = mel_lens[b];
  const size_t rowbase = (size_t)b * TT;

  const int  s      = tid;
  const bool active = s < SS;
  const bool isl    = (s & 1) != 0;        // odd state = label state
  const bool svalid = s < 2 * tl + 1;
  const int  col    = s >> 1;              // attn column for label states

  if (tid < 2) { alphaA[tid] = NEGF; alphaB[tid] = NEGF; }

  // Prologue: stage rows 0..PF-1 (800 lanes x b128) and the whole logZ row
  // (500 lanes x b128) asynchronously, then retire everything once.
  if (tid < PF * 100) {
    int r = tid / 100, j = tid - r * 100;
    async_cp16(attn + (rowbase + r) * LL + j * 4, &rowbuf[r][j * 4]);
  }
  if (tid < TT / 4) {
    async_cp16(logz + rowbase + tid * 4, &zbuf[tid * 4]);
  }
  wait_async<0>();
  __syncthreads();

  // alpha at t=0
  float a_self = NEGF;
  if (active) {
    float x  = isl ? rowbuf[0][col] : BLANK_LP;
    float e0 = svalid ? (x - zbuf[0]) : NEGF;
    a_self   = (s < 2) ? e0 : NEGF;
    alphaA[s + 2] = a_self;
  }
  __syncthreads();

  float* cur = alphaA;
  float* nxt = alphaB;
  for (int t = 1; t < ml; ++t) {
    // Issue prefetch of row t+PF-1 into the slot whose row (t-1) was fully
    // consumed before last step's barrier.
    int pr = t + (PF - 1);
    if (pr < ml && tid < 100) {
      async_cp16(attn + (rowbase + pr) * LL + tid * 4,
                 &rowbuf[pr & (PF - 1)][tid * 4]);
    }

    if (active) {
      float a1 = cur[s + 1];                  // alpha[s-1] (pad -> NEG at s=0)
      float a2 = isl ? cur[s] : NEGF;         // alpha[s-2] only for label states
      float m  = fmaxf(a_self, fmaxf(a1, a2));
      float sm = __expf(a_self - m) + __expf(a1 - m) + __expf(a2 - m);
      float x  = isl ? rowbuf[t & (PF - 1)][col] : BLANK_LP;
      float e  = svalid ? (x - zbuf[t]) : NEGF;
      a_self   = e + m + __logf(sm);
      nxt[s + 2] = a_self;
    }

    wait_async<PF - 2>();   // guarantees row t+1 is resident before the barrier
    __syncthreads();
    float* tmp = cur; cur = nxt; nxt = tmp;
  }

  if (tid == 0) {
    float aE = cur[2 * tl + 2];       // alpha[2*tl]
    float aO = cur[2 * tl + 1];       // alpha[2*tl - 1]
    float mm = fmaxf(aE, aO);
    float la = mm + __logf(__expf(aE - mm) + __expf(aO - mm));
    loss_b[b] = (la > NEGF * 0.5f) ? (-la / (float)tl) : 0.0f;
  }
}

// ---------- Kernel 3: mean over batch ---------------------------------------
__global__ void mean_kernel(const float* __restrict__ loss_b,
                            float* __restrict__ out, int n) {
  __shared__ float sh[64];
  int t = threadIdx.x;
  sh[t] = (t < n) ? loss_b[t] : 0.0f;
  __syncthreads();
#pragma unroll
  for (int o = 32; o >= 1; o >>= 1) {
    if (t < o) sh[t] += sh[t + o];
    __syncthreads();
  }
  if (t == 0) out[0] = sh[0] / (float)n;
}

extern "C" void kernel_launch(void* const* d_in, const int* in_sizes, int n_in,
                              void* d_out, int out_size, void* d_ws, size_t ws_size,
                              hipStream_t stream) {
  const float* attn = (const float*)d_in[0];
  const int*   tl   = (const int*)d_in[1];
  const int*   ml   = (const int*)d_in[2];
  const int    B    = in_sizes[1];          // 64

  float* logz   = (float*)d_ws;             // B*TT floats
  float* loss_b = logz + (size_t)B * TT;    // B floats
  int nrows = B * TT;

  logz_kernel<<<(nrows + 7) / 8, 256, 0, stream>>>(attn, logz, nrows);
  ctc_scan_kernel<<<B, NTH, 0, stream>>>(attn, tl, ml, logz, loss_b);
  mean_kernel<<<1, 64, 0, stream>>>(loss_b, (float*)d_out, B);
}